// GlobalNode_3418793967965
// MI455X (gfx1250) — compile-verified
//
#include <hip/hip_runtime.h>
#include <math.h>

typedef __attribute__((ext_vector_type(2))) float v2f;
typedef __attribute__((ext_vector_type(8))) float v8f;

#define EMB 128

// ---------------------------------------------------------------------------
// Helpers
// ---------------------------------------------------------------------------
__device__ __forceinline__ int lower_bound_i32(const int* __restrict__ a, int n, int key) {
    int lo = 0, hi = n;
    while (lo < hi) {
        int mid = (lo + hi) >> 1;
        if (a[mid] < key) lo = mid + 1; else hi = mid;
    }
    return lo;
}

// full wave32 all-reduce sum (every lane ends with the total)
__device__ __forceinline__ float wave_allreduce_sum(float s) {
#pragma unroll
    for (int m = 16; m >= 1; m >>= 1) s += __shfl_xor(s, m, 32);
    return s;
}

// ---------------------------------------------------------------------------
// Kernel 1: per-segment attentional pooling of x -> S[b,:] = sum_i alpha_i x_i
// One block per segment (batch_idx is sorted). 256 threads = 8 wave32.
// ---------------------------------------------------------------------------
__global__ __launch_bounds__(256, 1)
void seg_attn_pool_kernel(const float* __restrict__ x,
                          const int*   __restrict__ bidx,
                          const float* __restrict__ Wg,
                          const float* __restrict__ bg,
                          float* __restrict__ S,
                          float* __restrict__ flags,
                          int N)
{
    const int b    = blockIdx.x;
    const int tid  = threadIdx.x;
    const int wave = tid >> 5;
    const int lane = tid & 31;

    __shared__ float red[8];
    __shared__ float echunk[64];
    __shared__ float colbuf[EMB];

    const int s0  = lower_bound_i32(bidx, N, b);
    const int s1  = lower_bound_i32(bidx, N, b + 1);
    const int len = s1 - s0;

    if (len == 0) {
        if (tid < EMB) S[(size_t)b * EMB + tid] = 0.0f;
        if (tid == 0)  flags[b] = 0.0f;
        return;
    }

    // per-lane slice of gate weights (128 floats -> float4 per lane)
    const float4 wg4 = ((const float4*)Wg)[lane];
    const float  bgv = bg[0];

    // ---- Pass 1: segment max of gate = x . Wg + bg (one node per wave) ----
    float lmax = -INFINITY;
    for (int node = s0 + wave; node < s1; node += 8) {
        float4 xv = ((const float4*)(x + (size_t)node * EMB))[lane];
        float  s  = xv.x * wg4.x + xv.y * wg4.y + xv.z * wg4.z + xv.w * wg4.w;
        s = wave_allreduce_sum(s) + bgv;
        lmax = fmaxf(lmax, s);
    }
    if (lane == 0) red[wave] = lmax;
    __syncthreads();
    float gmax = red[0];
#pragma unroll
    for (int w = 1; w < 8; ++w) gmax = fmaxf(gmax, red[w]);
    __syncthreads();

    // ---- Pass 2: unnormalized weighted sum, chunked through LDS ----
    float dden = 0.0f;                 // wave-uniform partial denominator
    float acc  = 0.0f;                 // per-thread column accumulator
    const int col = tid & (EMB - 1);   // column 0..127
    const int par = tid >> 7;          // node parity 0/1

    for (int cb = 0; cb < len; cb += 64) {
        const int cnt = min(64, len - cb);
        // step a: recompute gates, e = exp(g - max)
        for (int j = wave; j < cnt; j += 8) {
            const int node = s0 + cb + j;
            float4 xv = ((const float4*)(x + (size_t)node * EMB))[lane];
            float  s  = xv.x * wg4.x + xv.y * wg4.y + xv.z * wg4.z + xv.w * wg4.w;
            s = wave_allreduce_sum(s) + bgv;
            float e = expf(s - gmax);
            dden += e;                 // same value in all lanes of the wave
            if (lane == 0) echunk[j] = e;
        }
        __syncthreads();
        // step b: acc[col] += e_j * x[node_j, col]  (coalesced, L2-hot re-read)
        for (int j = par; j < cnt; j += 2) {
            acc += echunk[j] * x[(size_t)(s0 + cb + j) * EMB + col];
        }
        __syncthreads();
    }

    if (lane == 0) red[wave] = dden;
    __syncthreads();
    float denom = 0.0f;
#pragma unroll
    for (int w = 0; w < 8; ++w) denom += red[w];
    const float inv = 1.0f / denom;    // denom >= 1 (max element contributes 1)

    if (par == 1) colbuf[col] = acc;
    __syncthreads();
    if (par == 0) S[(size_t)b * EMB + col] = (acc + colbuf[col]) * inv;
    if (tid == 0) flags[b] = 1.0f;
}

// ---------------------------------------------------------------------------
// Kernel 2: g = S @ Wv + flag*bv ; h = [g, g_prev] @ Wm + bm ;
//           out = g_prev + leaky_relu(h, 0.01)
// One block per 16-row slab; wave w owns output columns [16w, 16w+16).
// Uses V_WMMA_F32_16X16X4_F32 (fp32 matrix pipe, wave32).
// ---------------------------------------------------------------------------
__global__ __launch_bounds__(256, 1)
void pooled_mlp_kernel(const float* __restrict__ S,
                       const float* __restrict__ flags,
                       const float* __restrict__ gprev,
                       const float* __restrict__ Wv,
                       const float* __restrict__ bv,
                       const float* __restrict__ Wm,
                       const float* __restrict__ bm,
                       float* __restrict__ out)
{
    const int tid   = threadIdx.x;
    const int wave  = tid >> 5;
    const int lane  = tid & 31;
    const int r0    = blockIdx.x * 16;     // row slab base
    const int n0    = wave * 16;           // column tile base for this wave
    const int mn    = lane & 15;           // A-frag row / B,C-frag col within tile
    const int kh    = (lane >> 4) << 1;    // K sub-offset: 0 (lanes 0-15) or 2 (16-31)
    const int mbase = (lane >> 4) << 3;    // C/D: lanes 0-15 -> M=r, 16-31 -> M=8+r

    __shared__ float sbuf[16 * EMB];       // S slab
    __shared__ float pbuf[16 * EMB];       // g_prev slab
    __shared__ float gbuf[16 * EMB];       // computed g slab

    for (int i = tid; i < 16 * EMB; i += 256) {
        sbuf[i] = S[(size_t)r0 * EMB + i];
        pbuf[i] = gprev[(size_t)r0 * EMB + i];
    }
    __syncthreads();

    // ---- Stage 1: g = S_slab @ Wv + flag*bv  (K = 128, 32 WMMA ops) ----
    v8f c = {};
#pragma unroll 4
    for (int k = 0; k < EMB; k += 4) {
        v2f a, bfr;
        a[0]   = sbuf[mn * EMB + k + kh];
        a[1]   = sbuf[mn * EMB + k + kh + 1];
        bfr[0] = Wv[(k + kh) * EMB + n0 + mn];
        bfr[1] = Wv[(k + kh + 1) * EMB + n0 + mn];
        c = __builtin_amdgcn_wmma_f32_16x16x4_f32(false, a, false, bfr,
                                                  (short)0, c, false, false);
    }
#pragma unroll
    for (int r = 0; r < 8; ++r) {
        const int m = r + mbase;
        const int n = n0 + mn;
        const float fl = flags[r0 + m];    // 0 for empty segment (suppresses bv)
        gbuf[m * EMB + n] = c[r] + fl * bv[n];
    }
    __syncthreads();

    // ---- Stage 2: h = [g, g_prev] @ Wm + bm  (K = 256, 64 WMMA ops) ----
    v8f c2 = {};
#pragma unroll 4
    for (int k = 0; k < 2 * EMB; k += 4) {
        v2f a, bfr;
        const int ka = k + kh;             // ka, ka+1 always on same side of 128
        if (ka < EMB) {
            a[0] = gbuf[mn * EMB + ka];
            a[1] = gbuf[mn * EMB + ka + 1];
        } else {
            a[0] = pbuf[mn * EMB + ka - EMB];
            a[1] = pbuf[mn * EMB + ka - EMB + 1];
        }
        bfr[0] = Wm[ka * EMB + n0 + mn];
        bfr[1] = Wm[(ka + 1) * EMB + n0 + mn];
        c2 = __builtin_amdgcn_wmma_f32_16x16x4_f32(false, a, false, bfr,
                                                   (short)0, c2, false, false);
    }

    // ---- Epilogue: residual + leaky_relu ----
#pragma unroll
    for (int r = 0; r < 8; ++r) {
        const int m = r + mbase;
        const int n = n0 + mn;
        const float h  = c2[r] + bm[n];
        const float lr = h > 0.0f ? h : 0.01f * h;
        out[(size_t)(r0 + m) * EMB + n] = pbuf[m * EMB + n] + lr;
    }
}

// ---------------------------------------------------------------------------
// Host launcher
// ---------------------------------------------------------------------------
extern "C" void kernel_launch(void* const* d_in, const int* in_sizes, int n_in,
                              void* d_out, int out_size, void* d_ws, size_t ws_size,
                              hipStream_t stream)
{
    const float* x     = (const float*)d_in[0];
    const float* gprev = (const float*)d_in[1];
    const int*   bidx  = (const int*)  d_in[2];
    const float* Wg    = (const float*)d_in[3];
    const float* bg    = (const float*)d_in[4];
    const float* Wv    = (const float*)d_in[5];
    const float* bv    = (const float*)d_in[6];
    const float* Wm    = (const float*)d_in[7];
    const float* bm    = (const float*)d_in[8];
    float* out = (float*)d_out;

    const int N = in_sizes[0] / EMB;   // 524288
    const int B = in_sizes[1] / EMB;   // 1024

    float* S     = (float*)d_ws;               // [B, EMB]
    float* flags = S + (size_t)B * EMB;        // [B]

    seg_attn_pool_kernel<<<B, 256, 0, stream>>>(x, bidx, Wg, bg, S, flags, N);
    pooled_mlp_kernel<<<B / 16, 256, 0, stream>>>(S, flags, gprev, Wv, bv, Wm, bm, out);
}